// SignatureEncoder_31928786879179
// MI455X (gfx1250) — compile-verified
//
#include <hip/hip_runtime.h>

typedef float v2f __attribute__((ext_vector_type(2)));
typedef float v8f __attribute__((ext_vector_type(8)));

#define T_STEPS 1000
#define C_IN    18
#define C_AUG   19                            // time channel + 18 data channels
#define N_INC   (T_STEPS - 1)                 // 999 increments
#define OUT_PER_PATH (C_AUG + C_AUG * C_AUG)  // 19 + 361 = 380
#define K_MAIN  996                           // 249 full chunks of 4 increments

__device__ __forceinline__ v8f wmma_f32_4(v2f a, v2f b, v8f c) {
  // 8 args: (neg_a, A, neg_b, B, c_mod, C, reuse_a, reuse_b)
  return __builtin_amdgcn_wmma_f32_16x16x4_f32(false, a, false, b, (short)0, c,
                                               false, false);
}

__device__ __forceinline__ void store_tile(float* __restrict__ op, v8f a,
                                           int mt, int nt, int half, int ll) {
#pragma unroll
  for (int v = 0; v < 8; ++v) {
    int i = mt * 16 + v + 8 * half;  // C/D layout: lanes 16-31 hold M+8
    int j = nt * 16 + ll;
    if (i < C_AUG && j < C_AUG) op[C_AUG + i * C_AUG + j] = a[v];
  }
}

__global__ __launch_bounds__(256) void sig2_kernel(const float* __restrict__ x,
                                                   float* __restrict__ out,
                                                   int n_paths) {
  const int lane = threadIdx.x & 31;
  const int path = blockIdx.x * (blockDim.x >> 5) + (threadIdx.x >> 5);
  if (path >= n_paths) return;  // wave-uniform (grid divides exactly)

  const int half = lane >> 4;   // 0: K rows {0,1}   1: K rows {2,3}
  const int ll   = lane & 15;
  const int klo  = half * 2;
  const float inv = 1.0f / (float)N_INC;

  const float* __restrict__ xp = x + (size_t)path * (T_STEPS * C_IN);
  float* __restrict__ op = out + (size_t)path * OUT_PER_PATH;

  // tile 0: augmented channels 0..15 (0 = analytic time channel)
  // tile 1: augmented channels 16..18, rest padded with zeros
  const bool is_time  = (ll == 0);
  const bool t1_valid = (16 + ll) < C_AUG;
  const int  mc0 = is_time  ? 0 : (ll - 1);   // mem channel 0..14 (clamped)
  const int  mc1 = t1_valid ? (15 + ll) : 0;  // mem channel 15..17 (clamped)

  const float x0_0 = is_time  ? 0.0f : xp[mc0];  // x[k=0] per channel
  const float x0_1 = t1_valid ? xp[mc1] : 0.0f;

  const float* p0 = xp + klo * C_IN + mc0;  // base-row pointer, tile 0 channel
  const float* p1 = xp + klo * C_IN + mc1;  // base-row pointer, tile 1 channel

  v8f acc00 = {}, acc01 = {}, acc10 = {}, acc11 = {};

  // Running analytic time value at this lane's base row (row = k0 + klo).
  float trow = (float)klo * inv;
  const float t4 = 4.0f * inv;

  // ---- main loop: 249 chunks of 4 increments, no tail masks ----
  for (int it = 0; it < K_MAIN / 4; ++it) {
    float a0 = p0[0];
    float a1 = p0[C_IN];
    float a2 = p0[2 * C_IN];
    float b0 = p1[0];
    float b1 = p1[C_IN];
    float b2 = p1[2 * C_IN];
    // register-register selects only -> v_cndmask, no exec branching
    a0 = is_time ? trow            : a0;
    a1 = is_time ? trow + inv      : a1;
    a2 = is_time ? trow + 2.0f*inv : a2;
    b0 = t1_valid ? b0 : 0.0f;
    b1 = t1_valid ? b1 : 0.0f;
    b2 = t1_valid ? b2 : 0.0f;

    v2f aT0 = {0.5f * (a0 + a1) - x0_0, 0.5f * (a1 + a2) - x0_0};
    v2f aT1 = {0.5f * (b0 + b1) - x0_1, 0.5f * (b1 + b2) - x0_1};
    v2f bT0 = {a1 - a0, a2 - a1};
    v2f bT1 = {b1 - b0, b2 - b1};

    acc00 = wmma_f32_4(aT0, bT0, acc00);
    acc01 = wmma_f32_4(aT0, bT1, acc01);
    acc10 = wmma_f32_4(aT1, bT0, acc10);
    acc11 = wmma_f32_4(aT1, bT1, acc11);

    p0 += 4 * C_IN;
    p1 += 4 * C_IN;
    trow += t4;
  }

  // ---- epilogue chunk: increments 996..998 (row = 996 + klo) ----
  {
    const int row = K_MAIN + klo;              // 996 or 998
    int r2 = row + 2; if (r2 > T_STEPS - 1) r2 = T_STEPS - 1;
    const int d2 = (r2 - row) * C_IN;          // clamped 2-row offset
    float a0 = p0[0];
    float a1 = p0[C_IN];
    float a2 = p0[d2];
    float b0 = p1[0];
    float b1 = p1[C_IN];
    float b2 = p1[d2];
    a0 = is_time ? trow            : a0;
    a1 = is_time ? trow + inv      : a1;
    a2 = is_time ? trow + 2.0f*inv : a2;
    b0 = t1_valid ? b0 : 0.0f;
    b1 = t1_valid ? b1 : 0.0f;
    b2 = t1_valid ? b2 : 0.0f;

    const bool v1 = (row + 1) < N_INC;         // second increment of pair valid?
    v2f aT0, aT1, bT0, bT1;
    aT0.x = 0.5f * (a0 + a1) - x0_0;
    aT0.y = v1 ? (0.5f * (a1 + a2) - x0_0) : 0.0f;
    aT1.x = 0.5f * (b0 + b1) - x0_1;
    aT1.y = v1 ? (0.5f * (b1 + b2) - x0_1) : 0.0f;
    bT0.x = a1 - a0;
    bT0.y = v1 ? (a2 - a1) : 0.0f;
    bT1.x = b1 - b0;
    bT1.y = v1 ? (b2 - b1) : 0.0f;

    acc00 = wmma_f32_4(aT0, bT0, acc00);
    acc01 = wmma_f32_4(aT0, bT1, acc01);
    acc10 = wmma_f32_4(aT1, bT0, acc10);
    acc11 = wmma_f32_4(aT1, bT1, acc11);
  }

  // ---- S1: x[T-1] - x[0] (time channel contributes exactly 1.0) ----
  if (lane < C_AUG) {
    op[lane] = (lane == 0)
                   ? 1.0f
                   : (xp[(T_STEPS - 1) * C_IN + (lane - 1)] - xp[lane - 1]);
  }

  // ---- S2: scatter the 2x2 grid of 16x16 tiles into the 19x19 block ----
  store_tile(op, acc00, 0, 0, half, ll);
  store_tile(op, acc01, 0, 1, half, ll);
  store_tile(op, acc10, 1, 0, half, ll);
  store_tile(op, acc11, 1, 1, half, ll);
}

extern "C" void kernel_launch(void* const* d_in, const int* in_sizes, int n_in,
                              void* d_out, int out_size, void* d_ws, size_t ws_size,
                              hipStream_t stream) {
  const float* x = (const float*)d_in[0];
  float* out = (float*)d_out;
  const int n_paths = in_sizes[0] / (T_STEPS * C_IN);  // 128*16 = 2048
  const int waves_per_block = 8;                       // 256 threads
  dim3 block(32 * waves_per_block);
  dim3 grid((n_paths + waves_per_block - 1) / waves_per_block);
  sig2_kernel<<<grid, block, 0, stream>>>(x, out, n_paths);
}